// SortPoolingLayer_87127706567145
// MI455X (gfx1250) — compile-verified
//
#include <hip/hip_runtime.h>
#include <hip/hip_bf16.h>
#include <stdint.h>

#define KSEL 30
#define EMB_D 128
#define MAXSZ 140

typedef uint32_t uint32x4 __attribute__((ext_vector_type(4)));
typedef int32_t  int32x8  __attribute__((ext_vector_type(8)));

// ---- CDNA5 async global<->LDS DMA helpers (ASYNCcnt-tracked) ----
__device__ __forceinline__ void async_load_lds_b128(uint32_t lds_addr, const void* gptr) {
    asm volatile("global_load_async_to_lds_b128 %0, %1, off"
                 :: "v"(lds_addr), "v"((uint64_t)(uintptr_t)gptr) : "memory");
}
__device__ __forceinline__ void async_store_lds_b128(void* gptr, uint32_t lds_addr) {
    asm volatile("global_store_async_from_lds_b128 %0, %1, off"
                 :: "v"((uint64_t)(uintptr_t)gptr), "v"(lds_addr) : "memory");
}
__device__ __forceinline__ void wait_async0() {
    asm volatile("s_wait_asynccnt 0" ::: "memory");
}

// ---- CDNA5 Tensor Data Mover: strided 2-D column tile -> LDS ----
// D# group0/group1 built per CDNA5 ISA.
// 2-group form (VADDR2/VADDR3 = NULL) is legal for tensors up to 2-D.
__device__ __forceinline__ void tdm_load_column(uint32_t lds_addr,
                                                const void* gbase,
                                                int nrows, int stride_elems) {
    const uint64_t ga = (uint64_t)(uintptr_t)gbase;
    uint32x4 g0;
    g0.x = 1u;                                   // count=1 (user descriptor)
    g0.y = lds_addr;                             // LDS byte address
    g0.z = (uint32_t)ga;                         // global_addr[31:0]
    g0.w = (uint32_t)(ga >> 32) | (2u << 30);    // global_addr[56:32] | type=2

    int32x8 g1;
    g1[0] = (2 << 16);                           // data_size=2 -> 4 bytes
    g1[1] = (1 << 16);                           // tensor_dim0[15:0] = 1
    g1[2] = ((nrows & 0xffff) << 16);            // tensor_dim1[15:0] = nrows
    g1[3] = (1 << 16);                           // tile_dim0 = 1
    g1[4] = (nrows & 0xffff);                    // tile_dim1 = nrows (tile_dim2=0)
    g1[5] = stride_elems;                        // tensor_dim0_stride[31:0]
    g1[6] = 0;                                   // stride hi / dim1_stride lo
    g1[7] = 0;                                   // dim1_stride hi

    asm volatile("tensor_load_to_lds %0, %1" :: "s"(g0), "s"(g1) : "memory");
}
__device__ __forceinline__ void wait_tensor0() {
    __builtin_amdgcn_s_wait_tensorcnt(0);
}

// Kernel A: exclusive prefix sum of graph sizes -> offsets (single block).
__global__ void sortpool_scan_kernel(const int* __restrict__ sizes,
                                     int* __restrict__ offs, int G) {
    __shared__ int partial[256];
    const int t = threadIdx.x;
    const int per = (G + 255) / 256;
    int s = 0;
    for (int i = 0; i < per; ++i) {
        int g = t * per + i;
        if (g < G) s += sizes[g];
    }
    partial[t] = s;
    __syncthreads();
    int pre = 0;
    for (int i = 0; i < t; ++i) pre += partial[i];
    int run = pre;
    for (int i = 0; i < per; ++i) {
        int g = t * per + i;
        if (g < G) { offs[g] = run; run += sizes[g]; }
    }
}

// Kernel B: one 256-thread block (8 wave32s) per graph.
__global__ void sortpool_topk_gather_kernel(const float* __restrict__ emb,
                                            const int* __restrict__ sizes,
                                            const int* __restrict__ offs,
                                            float* __restrict__ pooled,
                                            float* __restrict__ idx_out) {
    __shared__ __align__(16) float xs[MAXSZ];          // sort-channel staging
    __shared__ int sel[KSEL];                          // selected local indices
    __shared__ __align__(16) float rows[KSEL * EMB_D]; // 15360B row staging

    const int g      = blockIdx.x;
    const int tid    = threadIdx.x;
    const int offset = offs[g];
    const int size   = sizes[g];

    const uint32_t xs_lds   = (uint32_t)(uintptr_t)(void*)xs;
    const uint32_t rows_lds = (uint32_t)(uintptr_t)(void*)rows;

    // ---- Phase 1: TDM DMA of the strided sort channel (last column) ----
    // Column tile: element i at emb[(offset+i)*128 + 127], i in [0,size).
    if (tid == 0) {
        const float* col = emb + (size_t)offset * EMB_D + (EMB_D - 1);
        tdm_load_column(xs_lds, col, size, EMB_D);
    }
    wait_tensor0();       // every wave drains TENSORcnt (no-op if it issued none)
    __syncthreads();      // make TDM's LDS writes visible to all waves

    // ---- Phase 2: rank-based stable top-K selection ----
    // rank(t) = #{ j : xs[j] > xs[t]  or  (xs[j] == xs[t] and j < t) }
    // Unique per element; rank < K  <=>  element is in top-K at position rank.
    if (tid < size) {
        const float v = xs[tid];
        int rank = 0;
        for (int j = 0; j < size; ++j) {
            float u = xs[j];
            rank += (u > v) || (u == v && j < tid);
        }
        if (rank < KSEL) sel[rank] = tid;
    }
    __syncthreads();

    // ---- Phase 3: emit indices (as float, per output layout) ----
    if (tid < KSEL) {
        idx_out[(size_t)g * KSEL + tid] = (float)(offset + sel[tid]);
    }

    // ---- Phase 4: gather 30 rows x 512B via async DMA through LDS ----
    const int wave = tid >> 5;
    const int lane = tid & 31;

    for (int k = wave; k < KSEL; k += 8) {
        const float* src = emb + (size_t)(offset + sel[k]) * EMB_D + lane * 4;
        async_load_lds_b128(rows_lds + (uint32_t)(k * EMB_D * 4 + lane * 16), src);
    }
    wait_async0();        // per-wave: its loads landed in LDS

    for (int k = wave; k < KSEL; k += 8) {
        float* dst = pooled + ((size_t)g * KSEL + k) * EMB_D + lane * 4;
        async_store_lds_b128(dst, rows_lds + (uint32_t)(k * EMB_D * 4 + lane * 16));
    }
    wait_async0();        // drain stores before wave retires
}

extern "C" void kernel_launch(void* const* d_in, const int* in_sizes, int n_in,
                              void* d_out, int out_size, void* d_ws, size_t ws_size,
                              hipStream_t stream) {
    const float* emb   = (const float*)d_in[0];
    const int*   sizes = (const int*)d_in[1];
    const int G = in_sizes[1];

    int* offs = (int*)d_ws;                             // G ints of scratch
    float* pooled  = (float*)d_out;                     // [G, K, D]
    float* idx_out = pooled + (size_t)G * KSEL * EMB_D; // [G, K]

    sortpool_scan_kernel<<<1, 256, 0, stream>>>(sizes, offs, G);
    sortpool_topk_gather_kernel<<<G, 256, 0, stream>>>(emb, sizes, offs, pooled, idx_out);
}